// FPthenAllConcatDecoder256_5231270166869
// MI455X (gfx1250) — compile-verified
//
#include <hip/hip_runtime.h>
#include <hip/hip_bf16.h>
#include <math.h>

// ---------------------------------------------------------------------------
// Types for WMMA f32 16x16x4 (CDNA5 / gfx1250, wave32)
// ---------------------------------------------------------------------------
typedef float v2f __attribute__((ext_vector_type(2)));
typedef float v8f __attribute__((ext_vector_type(8)));

#define BATCH 8
#define KCHUNK 32   // divides every I used here: 32,64,96,128,192,256,384,512,736,768

// ---------------------------------------------------------------------------
// GEMM + scale + bias + ReLU via V_WMMA_F32_16X16X4_F32, LDS-staged B operand.
// Y[b, coOff+o, n] = relu(S[o] * sum_i W[o,i] * X[b,i,n] + Bias[o])
// W: (O,I) row-major, X: (B,I,N), Y rows have stride strideOut (channels).
// Block = one 16-wide N tile; each wave owns one 16-row O tile and shares the
// X K-panel through LDS. Requires: O%16==0, I%KCHUNK==0, N%16==0.
// ---------------------------------------------------------------------------
__global__ void gemm_cb_wmma(const float* __restrict__ W,
                             const float* __restrict__ S,
                             const float* __restrict__ Bias,
                             const float* __restrict__ X,
                             float* __restrict__ Y,
                             int O, int I, int N, int strideOut, int coOff)
{
    __shared__ float xs[KCHUNK * 16];

    const int lane   = threadIdx.x & 31;
    const int wv     = threadIdx.x >> 5;
    const int nWaves = blockDim.x >> 5;
    const int tilesO = O >> 4;
    const int tn     = blockIdx.x;                    // N tile (block-uniform)
    const int to     = blockIdx.y * nWaves + wv;      // O tile (wave-uniform)
    const int b      = blockIdx.z;
    const bool active = (to < tilesO);                // wave-uniform

    const int m  = lane & 15;                         // O-row / N-col within tile
    const int kh = (lane >> 4) << 1;                  // 0 (lanes 0-15) / 2 (16-31)

    // A: 16x4 tile of W. lane m holds W[to*16+m, k+kh], W[.., k+kh+1]
    const float2* __restrict__ Arow = (const float2*)
        (W + (size_t)((active ? to : 0) * 16 + m) * I + kh);
    // X panel for this (b, N-tile): row k starts at Xpanel + k*N
    const float* __restrict__ Xpanel = X + (size_t)b * I * N + (size_t)tn * 16;

    v8f acc = {};
    for (int k0 = 0; k0 < I; k0 += KCHUNK) {
        // cooperative load of KCHUNK x 16 floats into LDS (b128 per lane)
        for (int idx = threadIdx.x; idx < KCHUNK * 4; idx += blockDim.x) {
            const int r = idx >> 2, q = idx & 3;
            *(float4*)&xs[r * 16 + q * 4] =
                *(const float4*)(Xpanel + (size_t)(k0 + r) * N + q * 4);
        }
        __syncthreads();
        if (active) {
#pragma unroll
            for (int ks = 0; ks < KCHUNK / 4; ++ks) {
                float2 a2 = Arow[(k0 >> 1) + ks * 2];     // W[m, k0+4ks+kh .. +1]
                v2f a; a.x = a2.x; a.y = a2.y;
                const float* xp = &xs[(ks * 4 + kh) * 16 + m];
                v2f bb; bb.x = xp[0]; bb.y = xp[16];      // X[k+kh, n], X[k+kh+1, n]
                acc = __builtin_amdgcn_wmma_f32_16x16x4_f32(
                    /*neg_a=*/false, a, /*neg_b=*/false, bb,
                    /*c_mod=*/(short)0, acc, /*reuse_a=*/false, /*reuse_b=*/false);
            }
        }
        __syncthreads();
    }

    if (active) {
        // D layout: VGPR v -> M=v (lanes 0-15), M=v+8 (lanes 16-31); N = lane&15
        const int oBase = to * 16 + ((lane >> 4) << 3);
        const int n = tn * 16 + m;
#pragma unroll
        for (int v = 0; v < 8; ++v) {
            const int o = oBase + v;
            float val = S[o] * acc[v] + Bias[o];
            val = fmaxf(val, 0.0f);
            Y[((size_t)b * strideOut + coOff + o) * N + n] = val;
        }
    }
}

// ---------------------------------------------------------------------------
// Brute-force 3-NN: for each dst point, find 3 nearest src points (squared
// distance, ascending) and write idx + inverse-distance weights.
// pD: (B,N,3)  pS: (B,M,3)  idxOut/wOut: (B,N,3)
// ---------------------------------------------------------------------------
__global__ void knn3_kernel(const float* __restrict__ pD,
                            const float* __restrict__ pS,
                            int N, int M,
                            int* __restrict__ idxOut,
                            float* __restrict__ wOut)
{
    __shared__ float sp[256 * 3];
    const int b = blockIdx.y;
    const int n = blockIdx.x * blockDim.x + threadIdx.x;
    const bool valid = (n < N);

    float px = 0.f, py = 0.f, pz = 0.f;
    if (valid) {
        const float* p = pD + ((size_t)b * N + n) * 3;
        px = p[0]; py = p[1]; pz = p[2];
    }
    float d0 = 3.4e38f, d1 = 3.4e38f, d2 = 3.4e38f;
    int   i0 = 0,       i1 = 0,       i2 = 0;

    for (int base = 0; base < M; base += 256) {
        const int cnt = min(256, M - base);
        for (int t = threadIdx.x; t < cnt * 3; t += blockDim.x)
            sp[t] = pS[((size_t)b * M + base) * 3 + t];
        __syncthreads();
        if (valid) {
            for (int j = 0; j < cnt; ++j) {
                const float dx = px - sp[j * 3 + 0];
                const float dy = py - sp[j * 3 + 1];
                const float dz = pz - sp[j * 3 + 2];
                const float d = dx * dx + dy * dy + dz * dz;
                const int gi = base + j;
                if (d < d0)      { d2 = d1; i2 = i1; d1 = d0; i1 = i0; d0 = d; i0 = gi; }
                else if (d < d1) { d2 = d1; i2 = i1; d1 = d;  i1 = gi; }
                else if (d < d2) { d2 = d;  i2 = gi; }
            }
        }
        __syncthreads();
    }

    if (valid) {
        const float r0 = 1.0f / (sqrtf(fmaxf(d0, 0.0f)) + 1e-8f);
        const float r1 = 1.0f / (sqrtf(fmaxf(d1, 0.0f)) + 1e-8f);
        const float r2 = 1.0f / (sqrtf(fmaxf(d2, 0.0f)) + 1e-8f);
        const float rs = r0 + r1 + r2;
        const size_t o = ((size_t)b * N + n) * 3;
        idxOut[o + 0] = i0; idxOut[o + 1] = i1; idxOut[o + 2] = i2;
        wOut[o + 0] = r0 / rs; wOut[o + 1] = r1 / rs; wOut[o + 2] = r2 / rs;
    }
}

// ---------------------------------------------------------------------------
// Weighted gather: out[b, coOff+c, n] = sum_k w[b,n,k] * fS[b, c, idx[b,n,k]]
// fS: (B,C,M)  out rows stride strideOut.
// ---------------------------------------------------------------------------
__global__ void interp_gather(const float* __restrict__ fS,
                              const int* __restrict__ idx,
                              const float* __restrict__ w,
                              float* __restrict__ out,
                              int C, int M, int N, int strideOut, int coOff,
                              size_t total)
{
    size_t tid = (size_t)blockIdx.x * blockDim.x + threadIdx.x;
    if (tid >= total) return;
    const int n = (int)(tid % N);
    size_t t = tid / N;
    const int c = (int)(t % C);
    const int b = (int)(t / C);

    const size_t kb = ((size_t)b * N + n) * 3;
    const int*   ip = idx + kb;
    const float* wp = w + kb;
    const float* f = fS + ((size_t)b * C + c) * M;
    const float v = wp[0] * f[ip[0]] + wp[1] * f[ip[1]] + wp[2] * f[ip[2]];
    out[((size_t)b * strideOut + coOff + c) * N + n] = v;
}

// ---------------------------------------------------------------------------
// Channel copy into a concat buffer: out[b, coOff+c, n] = in[b, c, n]
// ---------------------------------------------------------------------------
__global__ void copy_channels(const float* __restrict__ in,
                              float* __restrict__ out,
                              int C, int N, int strideOut, int coOff,
                              size_t total)
{
    size_t tid = (size_t)blockIdx.x * blockDim.x + threadIdx.x;
    if (tid >= total) return;
    const int n = (int)(tid % N);
    size_t t = tid / N;
    const int c = (int)(t % C);
    const int b = (int)(t / C);
    out[((size_t)b * strideOut + coOff + c) * N + n] =
        in[((size_t)b * C + c) * N + n];
}

// ---------------------------------------------------------------------------
// Host-side orchestration
// ---------------------------------------------------------------------------
extern "C" void kernel_launch(void* const* d_in, const int* in_sizes, int n_in,
                              void* d_out, int out_size, void* d_ws, size_t ws_size,
                              hipStream_t stream)
{
    (void)in_sizes; (void)n_in; (void)out_size; (void)ws_size;

    const int N1 = 8192, N2 = 2048, N3 = 512, N4 = 128, N5 = 32;

    const float* p1 = (const float*)d_in[0];
    const float* f1 = (const float*)d_in[1];
    const float* p2 = (const float*)d_in[2];
    const float* f2 = (const float*)d_in[3];
    const float* p3 = (const float*)d_in[4];
    const float* f3 = (const float*)d_in[5];
    const float* p4 = (const float*)d_in[6];
    const float* f4 = (const float*)d_in[7];
    const float* p5 = (const float*)d_in[8];
    const float* f5 = (const float*)d_in[9];

    struct Blk { const float *w, *s, *b; };
    auto blk = [&](int i) -> Blk {
        return Blk{ (const float*)d_in[i], (const float*)d_in[i + 1], (const float*)d_in[i + 2] };
    };
    // flattened params order: dec4[0],dec4[1],dec3[0],dec3[1],dec2[0],dec2[1],
    //                         dec1[0],dec1[1], t5,t4,t3,t2, fuse[0],fuse[1]
    Blk dec4a = blk(10), dec4b = blk(13);
    Blk dec3a = blk(16), dec3b = blk(19);
    Blk dec2a = blk(22), dec2b = blk(25);
    Blk dec1a = blk(28), dec1b = blk(31);
    Blk t5 = blk(34), t4 = blk(37), t3 = blk(40), t2 = blk(43);
    Blk fuse0 = blk(46), fuse1 = blk(49);

    // ------------------- workspace bump allocator -------------------
    float* wsf = (float*)d_ws;
    size_t off = 0;
    auto alloc = [&](size_t nFloats) -> float* {
        float* p = wsf + off;
        off += (nFloats + 63) & ~(size_t)63;
        return p;
    };

    float* h5    = alloc((size_t)BATCH * 256 * N5);   // t5(f5)
    float* cat4  = alloc((size_t)BATCH * 768 * N4);
    float* x4a   = alloc((size_t)BATCH * 256 * N4);
    float* f4n   = alloc((size_t)BATCH * 256 * N4);
    float* g4src = alloc((size_t)BATCH * 256 * N4);   // t4(f4n)
    float* cat3  = alloc((size_t)BATCH * 384 * N3);
    float* x3a   = alloc((size_t)BATCH * 128 * N3);
    float* f3n   = alloc((size_t)BATCH * 128 * N3);
    float* g3src = alloc((size_t)BATCH * 128 * N3);   // t3(f3n)
    float* cat2  = alloc((size_t)BATCH * 192 * N2);
    float* x2a   = alloc((size_t)BATCH * 64 * N2);
    float* f2n   = alloc((size_t)BATCH * 64 * N2);
    float* g2src = alloc((size_t)BATCH * 64 * N2);    // t2(f2n)
    float* cat1  = alloc((size_t)BATCH * 96 * N1);
    float* x1a   = alloc((size_t)BATCH * 32 * N1);
    float* fcat  = alloc((size_t)BATCH * 736 * N1);
    float* fmid  = alloc((size_t)BATCH * 64 * N1);
    float* w45   = alloc((size_t)BATCH * N4 * 3);
    int*   i45   = (int*)alloc((size_t)BATCH * N4 * 3);
    float* w34   = alloc((size_t)BATCH * N3 * 3);
    int*   i34   = (int*)alloc((size_t)BATCH * N3 * 3);
    float* w23   = alloc((size_t)BATCH * N2 * 3);
    int*   i23   = (int*)alloc((size_t)BATCH * N2 * 3);
    float* w1x   = alloc((size_t)BATCH * N1 * 3);     // reused for p1<-p2/p3/p4/p5
    int*   i1x   = (int*)alloc((size_t)BATCH * N1 * 3);

    // ------------------- launch helpers -------------------
    auto gemm = [&](Blk bk, const float* X, float* Y, int O, int I, int N,
                    int strideOut, int coOff) {
        const int tilesO = O >> 4;
        const int waves  = tilesO < 4 ? tilesO : 4;
        dim3 g(N >> 4, (tilesO + waves - 1) / waves, BATCH);
        gemm_cb_wmma<<<g, dim3(32 * waves), 0, stream>>>(bk.w, bk.s, bk.b, X, Y,
                                                         O, I, N, strideOut, coOff);
    };
    auto knn = [&](const float* pD, const float* pS, int N, int M,
                   int* idx, float* w) {
        dim3 g((N + 255) / 256, BATCH);
        knn3_kernel<<<g, 256, 0, stream>>>(pD, pS, N, M, idx, w);
    };
    auto interp = [&](const float* fS, const int* idx, const float* w,
                      float* out, int C, int M, int N, int so, int co) {
        size_t tot = (size_t)BATCH * C * N;
        interp_gather<<<(unsigned)((tot + 255) / 256), 256, 0, stream>>>(
            fS, idx, w, out, C, M, N, so, co, tot);
    };
    auto copych = [&](const float* in, float* out, int C, int N, int so, int co) {
        size_t tot = (size_t)BATCH * C * N;
        copy_channels<<<(unsigned)((tot + 255) / 256), 256, 0, stream>>>(
            in, out, C, N, so, co, tot);
    };

    // ------------------- pipeline -------------------
    // t5: f5 (B,512,32) -> h5 (B,256,32)
    gemm(t5, f5, h5, 256, 512, N5, 256, 0);

    // dec4: cat(f4, interp(p4,p5,f5)) -> 256 -> 256
    knn(p4, p5, N4, N5, i45, w45);
    copych(f4, cat4, 256, N4, 768, 0);
    interp(f5, i45, w45, cat4, 512, N5, N4, 768, 256);
    gemm(dec4a, cat4, x4a, 256, 768, N4, 256, 0);
    gemm(dec4b, x4a, f4n, 256, 256, N4, 256, 0);
    gemm(t4, f4n, g4src, 256, 256, N4, 256, 0);

    // dec3
    knn(p3, p4, N3, N4, i34, w34);
    copych(f3, cat3, 128, N3, 384, 0);
    interp(f4n, i34, w34, cat3, 256, N4, N3, 384, 128);
    gemm(dec3a, cat3, x3a, 128, 384, N3, 128, 0);
    gemm(dec3b, x3a, f3n, 128, 128, N3, 128, 0);
    gemm(t3, f3n, g3src, 128, 128, N3, 128, 0);

    // dec2
    knn(p2, p3, N2, N3, i23, w23);
    copych(f2, cat2, 64, N2, 192, 0);
    interp(f3n, i23, w23, cat2, 128, N3, N2, 192, 64);
    gemm(dec2a, cat2, x2a, 64, 192, N2, 64, 0);
    gemm(dec2b, x2a, f2n, 64, 64, N2, 64, 0);
    gemm(t2, f2n, g2src, 64, 64, N2, 64, 0);

    // dec1: cat(f1, interp(p1,p2,f2n)) -> 32 -> 32  (f1n goes to fcat[:,0:32,:])
    knn(p1, p2, N1, N2, i1x, w1x);
    copych(f1, cat1, 32, N1, 96, 0);
    interp(f2n, i1x, w1x, cat1, 64, N2, N1, 96, 32);
    gemm(dec1a, cat1, x1a, 32, 96, N1, 32, 0);
    gemm(dec1b, x1a, fcat, 32, 32, N1, 736, 0);

    // g2 = interp(p1,p2, t2(f2n)) -> fcat[:,32:96,:]   (reuses p1<-p2 knn)
    interp(g2src, i1x, w1x, fcat, 64, N2, N1, 736, 32);
    // g3 = interp(p1,p3, t3(f3n)) -> fcat[:,96:224,:]
    knn(p1, p3, N1, N3, i1x, w1x);
    interp(g3src, i1x, w1x, fcat, 128, N3, N1, 736, 96);
    // g4 = interp(p1,p4, t4(f4n)) -> fcat[:,224:480,:]
    knn(p1, p4, N1, N4, i1x, w1x);
    interp(g4src, i1x, w1x, fcat, 256, N4, N1, 736, 224);
    // g5 = interp(p1,p5, t5(f5)) -> fcat[:,480:736,:]
    knn(p1, p5, N1, N5, i1x, w1x);
    interp(h5, i1x, w1x, fcat, 256, N5, N1, 736, 480);

    // fuse: 736 -> 64 -> 32 (final straight into d_out, (B,32,8192))
    gemm(fuse0, fcat, fmid, 64, 736, N1, 64, 0);
    gemm(fuse1, fmid, (float*)d_out, 32, 64, N1, 32, 0);
}